// kernel_generated_2_21887153341262
// MI455X (gfx1250) — compile-verified
//
#include <hip/hip_runtime.h>

// CDNA5 / gfx1250 — fused shift-conv as one fp32 WMMA GEMM pass.
//
// GEMM: M=392 (p=(o,n,m)), N=256 (k), K=768 (q=i*3+j)
//   A[p,q] = x[o, i, n, (m+j-2) mod 14]   masked by (0 <= m+j-1 < 14)
//   B[q,k] = w[i, k, j]                   (w layout: i, k, j)
//   final:  out[o*256+k, (n+1) mod 14, m] = gemm[p,k]   (H-roll +1)
//
// q -> (i,j) is periodic with period 12 in q (= 3 K-steps of 4), so all
// division/modulo address math is hoisted: 6 A pointers + 6 B pointers are
// precomputed per lane and advanced by constants (4*196 / 4*768 floats) per
// period. B's 4 N-sub-tiles use instruction immediate offsets (0/192/384/576B).

typedef float v2f __attribute__((ext_vector_type(2)));
typedef float v8f __attribute__((ext_vector_type(8)));

__global__ __launch_bounds__(32)
void shiftconv_wmma_f32(const float* __restrict__ x,
                        const float* __restrict__ w,
                        float* __restrict__ out) {
  const int lane   = threadIdx.x;   // 0..31 (wave32)
  const int mt     = blockIdx.x;    // 0..24  M-tile (rows mt*16 .. mt*16+15)
  const int ng     = blockIdx.y;    // 0..3   N-group of 64 columns
  const int lanelo = lane & 15;
  const int hi     = lane >> 4;     // half-wave selects K pair {0,1} vs {2,3}

  // ---- A-side row geometry for this lane (A fragment: M = lanelo) ----
  const int  r  = mt * 16 + lanelo;
  const bool rv = (r < 392);
  const int  rr = rv ? r : 0;              // clamp padded rows to a valid addr
  const int  o   = rr / 196;
  const int  rem = rr - o * 196;
  const int  n   = rem / 14;
  const int  m   = rem - n * 14;
  const float* xb = x + (o * 256) * 196 + n * 14;  // + i*196 + wcol

  // Per-tap W column (always a valid address) and 0/1 mask factor.
  // tap j reads rolled W index (m+j-1); value = x[... (m+j-2) mod 14 ...]
  const float rvf = rv ? 1.0f : 0.0f;
  float af[3];
  int   wc[3];
  af[0] = (m >= 1)  ? rvf : 0.0f;  wc[0] = (m + 12) % 14;  // (m-2) mod 14
  af[1] = rvf;                     wc[1] = (m + 13) % 14;  // (m-1) mod 14
  af[2] = (m <= 12) ? rvf : 0.0f;  wc[2] = m;              //  m

  // ---- B-side: column base for sub-tile 0 of this lane ----
  const int kbase = ng * 64 + lanelo;      // k of sub-tile 0; +16/+32/+48 others

  // ---- Hoist all q->(i,j) math: 6 slots u = (t,v), t=K-step in period, v=elt
  const int qoff = hi * 2;                 // K-pair offset within a 4-wide step
  const float* pA[6];
  const float* pB[6];
  float        mA[6];
  #pragma unroll
  for (int u = 0; u < 6; ++u) {
    const int t = u >> 1, v = u & 1;
    const int q = t * 4 + qoff + v;        // 0..11, first period
    const int i = (int)((unsigned)q / 3u);
    const int j = q - i * 3;
    pA[u] = xb + i * 196 + wc[j];
    mA[u] = af[j];
    pB[u] = w + i * 768 + j + kbase * 3;   // + {0,48,96,144} floats per sub-tile
  }

  v8f acc0 = {}, acc1 = {}, acc2 = {}, acc3 = {};

  // 192 K-steps = 64 periods of 3; each period: 12 WMMAs, constant-stride bumps
  #pragma unroll 2
  for (int it = 0; it < 64; ++it) {
    #pragma unroll
    for (int t = 0; t < 3; ++t) {
      const int u0 = 2 * t, u1 = 2 * t + 1;

      v2f a;
      a.x = pA[u0][0] * mA[u0];
      a.y = pA[u1][0] * mA[u1];

      v2f b0, b1, b2, b3;                   // immediate-offset loads
      b0.x = pB[u0][  0];  b0.y = pB[u1][  0];
      b1.x = pB[u0][ 48];  b1.y = pB[u1][ 48];
      b2.x = pB[u0][ 96];  b2.y = pB[u1][ 96];
      b3.x = pB[u0][144];  b3.y = pB[u1][144];

      acc0 = __builtin_amdgcn_wmma_f32_16x16x4_f32(false, a, false, b0, (short)0, acc0, false, false);
      acc1 = __builtin_amdgcn_wmma_f32_16x16x4_f32(false, a, false, b1, (short)0, acc1, false, false);
      acc2 = __builtin_amdgcn_wmma_f32_16x16x4_f32(false, a, false, b2, (short)0, acc2, false, false);
      acc3 = __builtin_amdgcn_wmma_f32_16x16x4_f32(false, a, false, b3, (short)0, acc3, false, false);
    }
    #pragma unroll
    for (int u = 0; u < 6; ++u) {
      pA[u] += 4 * 196;                     // next 4 input channels
      pB[u] += 4 * 768;
    }
  }

  // ---- store: C/D layout -> VGPR d holds row (d + hi*8), col = lanelo ----
  #pragma unroll
  for (int d = 0; d < 8; ++d) {
    const int p = mt * 16 + hi * 8 + d;
    if (p < 392) {
      const int po   = p / 196;
      const int prem = p - po * 196;
      const int pn   = prem / 14;
      const int pm   = prem - pn * 14;
      const int pn1  = (pn + 1) % 14;        // final H-roll (+1)
      float* ob = out + (po * 256 + ng * 64 + lanelo) * 196 + pn1 * 14 + pm;
      ob[ 0 * 196] = acc0[d];
      ob[16 * 196] = acc1[d];
      ob[32 * 196] = acc2[d];
      ob[48 * 196] = acc3[d];
    }
  }
}

extern "C" void kernel_launch(void* const* d_in, const int* in_sizes, int n_in,
                              void* d_out, int out_size, void* d_ws, size_t ws_size,
                              hipStream_t stream) {
  const float* x = (const float*)d_in[0];   // (1,512,14,14)
  const float* w = (const float*)d_in[1];   // (256,256,3)
  float* out = (float*)d_out;               // (1,512,14,14)
  (void)in_sizes; (void)n_in; (void)out_size; (void)d_ws; (void)ws_size;

  dim3 grid(25, 4);   // 25 M-tiles x 4 N-groups (64 cols each)
  shiftconv_wmma_f32<<<grid, 32, 0, stream>>>(x, w, out);
}